// JointRegressionScale_85031762526747
// MI455X (gfx1250) — compile-verified
//
#include <hip/hip_runtime.h>
#include <hip/hip_bf16.h>
#include <math.h>

typedef float v2f __attribute__((ext_vector_type(2)));
typedef float v8f __attribute__((ext_vector_type(8)));

#define N_OBS     131072
#define P_DIM     64
#define C_HUB_C   1.345f
#define ALPHA_C   0.35514f
#define GAMMA_C   2.0f
#define MAXITER   1500
#define EPS_C     1e-5f
#define DENOM_C   (2.0f * ALPHA_C * (float)(N_OBS - P_DIM - 1))

#define GB_GRAM   128            // blocks in gram kernel (512 thr each)
#define GA_RES    512            // blocks in residual kernel (256 thr -> 1 row/thread)
#define GP_BLK    256            // blocks in pseudo-residual kernel
#define WAVES_PSEU (GP_BLK * 8)  // 2048 waves
#define ROWS_PER_WAVE (N_OBS / WAVES_PSEU) // 64

// ---- workspace layout (float offsets) ----
struct HubState {
  float scale, med1, med2, crit;
  unsigned prefix, kk, cnt, done;
  unsigned pad[8];
};
#define OFF_COUNTS  16                        // 256 uints (radix histogram)
#define OFF_CHI     288                       // 512 f (chi partials)
#define OFF_SPART   800                       // 2048 f (sum r_pseu partials)
#define OFF_BETA    2848                      // 64 f
#define OFF_G       2912                      // 4096 f
#define OFF_GINV    7008                      // 4096 f
#define OFF_AMAT    11104                     // 4096 f  (A = W * Ginv)
#define OFF_R       15200                     // 131072 f (residuals)
#define OFF_VPART   (OFF_R + N_OBS)           // 2048*64 f
#define OFF_GPART   (OFF_VPART + WAVES_PSEU*64) // 128*4096 f

__device__ __forceinline__ float new_scale(float scale, float Schi) {
  return sqrtf(GAMMA_C * scale * scale / DENOM_C * Schi);
}

// ---------------- init: reset all persistent state each call ----------------
__global__ void __launch_bounds__(256) k_init(float* ws) {
  HubState* st = (HubState*)ws;
  int t = threadIdx.x;
  if (t == 0) {
    st->scale = 0.f; st->med1 = 0.f; st->med2 = 0.f; st->crit = __builtin_inff();
    st->prefix = 0u; st->kk = (unsigned)((N_OBS - 1) / 2); st->cnt = 0u; st->done = 0u;
  }
  ((unsigned*)(ws + OFF_COUNTS))[t] = 0u;
  if (t < 64) ws[OFF_BETA + t] = 0.0f;
}

// ---------------- exact lower-median via 4-pass byte radix select ----------------
__global__ void __launch_bounds__(256) k_hist(const float* bvec, float* ws, int mode, int pass) {
  HubState* st = (HubState*)ws;
  unsigned* counts = (unsigned*)(ws + OFF_COUNTS);
  __shared__ unsigned sh[256];
  int t = threadIdx.x;
  sh[t] = 0u;
  __syncthreads();
  float med1 = st->med1;
  unsigned prefix = st->prefix;
  int shift = 24 - 8 * pass;
  for (int i = blockIdx.x * 256 + t; i < N_OBS; i += gridDim.x * 256) {
    float v = bvec[i];
    if (mode) v = fabsf(v - med1);
    unsigned u = __float_as_uint(v);
    unsigned key = (u & 0x80000000u) ? ~u : (u | 0x80000000u); // order-preserving map
    bool ok = (pass == 0) || ((key >> (shift + 8)) == prefix);
    if (ok) atomicAdd(&sh[(key >> shift) & 255u], 1u);
  }
  __syncthreads();
  if (sh[t]) atomicAdd(&counts[t], sh[t]);
}

__global__ void k_select(float* ws, int mode, int pass) {
  HubState* st = (HubState*)ws;
  unsigned* counts = (unsigned*)(ws + OFF_COUNTS);
  unsigned kk = st->kk, cum = 0u, bucket = 255u;
  for (unsigned b = 0; b < 256u; ++b) {
    unsigned c = counts[b];
    if (kk < cum + c) { bucket = b; kk -= cum; break; }
    cum += c;
  }
  st->prefix = (st->prefix << 8) | bucket;
  st->kk = kk;
  for (int b = 0; b < 256; ++b) counts[b] = 0u;
  if (pass == 3) {
    unsigned key = st->prefix;
    unsigned u = (key & 0x80000000u) ? (key ^ 0x80000000u) : ~key;
    float med = __uint_as_float(u);
    if (mode == 0) st->med1 = med; else st->med2 = med;
    st->prefix = 0u; st->kk = (unsigned)((N_OBS - 1) / 2);
  }
}

__global__ void k_scale0(float* ws) {
  HubState* st = (HubState*)ws;
  st->scale = 1.4815f * st->med2;   // 1.4815 * MAD (lower medians)
}

// ---------------- Gram matrix G = X^T X via fp32 WMMA 16x16x4 ----------------
// 16 waves per block, wave w owns output tile (I=w/4, J=w%4) of the 64x64 G.
// A-layout (16x4 f32): lane<16 {K0,K1}, lane>=16 {K2,K3}; B (4x16) symmetric.
__global__ void __launch_bounds__(512) k_gram(const float* __restrict__ X, float* ws) {
  float* Gpart = ws + OFF_GPART + (size_t)blockIdx.x * 4096;
  int wave = threadIdx.x >> 5, lane = threadIdx.x & 31;
  int I = wave >> 2, J = wave & 3;
  int g = lane >> 4, lm = lane & 15;
  int colA = 16 * I + lm, colB = 16 * J + lm;
  const int rows = N_OBS / GB_GRAM;        // 1024
  const int r0 = blockIdx.x * rows;
  v8f acc = {};
  for (int k = r0; k < r0 + rows; k += 4) {
    int rr = k + 2 * g;
    v2f a, b;
    a.x = X[(size_t)rr * 64 + colA];
    a.y = X[(size_t)(rr + 1) * 64 + colA];
    b.x = X[(size_t)rr * 64 + colB];
    b.y = X[(size_t)(rr + 1) * 64 + colB];
    if (k + 8 < r0 + rows)
      __builtin_prefetch(&X[(size_t)(rr + 8) * 64 + colA], 0, 1);
    acc = __builtin_amdgcn_wmma_f32_16x16x4_f32(
        /*neg_a=*/false, a, /*neg_b=*/false, b,
        /*c_mod=*/(short)0, acc, /*reuse_a=*/false, /*reuse_b=*/false);
  }
  #pragma unroll
  for (int r = 0; r < 8; ++r) {
    int m = r + 8 * g;                      // C/D layout: M = vgpr + 8*(lane/16)
    Gpart[(16 * I + m) * 64 + 16 * J + lm] = acc[r];
  }
}

__global__ void __launch_bounds__(256) k_gram_reduce(float* ws) {
  int idx = blockIdx.x * 256 + threadIdx.x;      // 0..4095
  float s = 0.f;
  const float* gp = ws + OFF_GPART;
  for (int p = 0; p < GB_GRAM; ++p) s += gp[(size_t)p * 4096 + idx];
  ws[OFF_G + idx] = s;
}

// ---------------- Gauss-Jordan inverse of SPD 64x64 in LDS ----------------
__global__ void __launch_bounds__(256) k_invert(float* ws) {
  __shared__ float aug[64][129];
  __shared__ float fac[64];
  __shared__ float pinv_s;
  int t = threadIdx.x;
  const float* G = ws + OFF_G;
  for (int idx = t; idx < 4096; idx += 256) {
    int r = idx >> 6, c = idx & 63;
    aug[r][c] = G[idx];
    aug[r][64 + c] = (r == c) ? 1.f : 0.f;
  }
  __syncthreads();
  for (int k = 0; k < 64; ++k) {
    if (t == 0) pinv_s = 1.0f / aug[k][k];
    __syncthreads();
    if (t < 128) aug[k][t] *= pinv_s;
    __syncthreads();
    if (t < 64) fac[t] = aug[t][k];
    __syncthreads();
    for (int idx = t; idx < 64 * 128; idx += 256) {
      int r = idx >> 7, c = idx & 127;
      if (r != k) aug[r][c] -= fac[r] * aug[k][c];
    }
    __syncthreads();
  }
  for (int idx = t; idx < 4096; idx += 256) {
    int r = idx >> 6, c = idx & 63;
    ws[OFF_GINV + idx] = aug[r][64 + c];
  }
}

// ---------------- A = W * Ginv (loop-invariant, 64x64x64) ----------------
__global__ void __launch_bounds__(256) k_prepA(const float* __restrict__ W, float* ws) {
  __shared__ float sW[64][65];
  __shared__ float sGi[64][65];
  int t = threadIdx.x;
  for (int idx = t; idx < 4096; idx += 256) {
    int r = idx >> 6, c = idx & 63;
    sW[r][c]  = W[idx];
    sGi[r][c] = ws[OFF_GINV + idx];
  }
  __syncthreads();
  for (int idx = t; idx < 4096; idx += 256) {
    int r = idx >> 6, c = idx & 63;
    float s = 0.f;
    #pragma unroll 8
    for (int m = 0; m < 64; ++m) s += sW[r][m] * sGi[m][c];
    ws[OFF_AMAT + idx] = s;
  }
}

// ---------------- iteration pass 1: r = b - X*beta ; chi partials ----------------
__global__ void __launch_bounds__(256) k_resid(const float* __restrict__ X,
                                               const float* __restrict__ bvec, float* ws) {
  HubState* st = (HubState*)ws;
  if (st->done) return;
  __shared__ float sb[64];
  __shared__ float sred[256];
  int t = threadIdx.x;
  if (t < 64) sb[t] = ws[OFF_BETA + t];
  __syncthreads();
  int row = blockIdx.x * 256 + t;               // exactly N threads total
  float scale = st->scale;
  const float4* xr = (const float4*)(X + (size_t)row * 64);
  float sum = 0.f;
  #pragma unroll
  for (int q = 0; q < 16; ++q) {
    float4 x = xr[q];
    sum += x.x * sb[4*q] + x.y * sb[4*q+1] + x.z * sb[4*q+2] + x.w * sb[4*q+3];
  }
  float rv = bvec[row] - sum;
  ws[OFF_R + row] = rv;
  float u = rv / scale;
  sred[t] = 0.5f * fminf(u * u, C_HUB_C * C_HUB_C);   // psi(u)*u - rho(u)
  __syncthreads();
  for (int s = 128; s > 0; s >>= 1) { if (t < s) sred[t] += sred[t + s]; __syncthreads(); }
  if (t == 0) ws[OFF_CHI + blockIdx.x] = sred[0];
}

// ------- pass 2: recompute scale locally; r_pseu; v = X^T r_pseu partials -------
__global__ void __launch_bounds__(256) k_pseu(const float* __restrict__ X, float* ws) {
  HubState* st = (HubState*)ws;
  if (st->done) return;
  float scale = st->scale;
  float Schi = 0.f;
  const float* chi = ws + OFF_CHI;
  for (int i = 0; i < GA_RES; ++i) Schi += chi[i];   // identical order everywhere
  float ns = new_scale(scale, Schi);
  float cns = C_HUB_C * ns;                          // psi(r/ns)*ns = clamp(r, +-c*ns)
  int wave = threadIdx.x >> 5, lane = threadIdx.x & 31;
  int wg = blockIdx.x * 8 + wave;
  int base = wg * ROWS_PER_WAVE;
  float acc0 = 0.f, acc1 = 0.f, sacc = 0.f;
  const float* rr = ws + OFF_R;
  for (int m = 0; m < ROWS_PER_WAVE; ++m) {
    int i = base + m;
    float rv = rr[i];                                // broadcast load
    float rp = fminf(fmaxf(rv, -cns), cns);
    sacc += rp;
    v2f xv = ((const v2f*)(X + (size_t)i * 64))[lane]; // lanes cover 64 cols, coalesced
    acc0 += xv.x * rp;
    acc1 += xv.y * rp;
  }
  float* vp = ws + OFF_VPART + (size_t)wg * 64;
  vp[2 * lane]     = acc0;
  vp[2 * lane + 1] = acc1;
  if (lane == 0) ws[OFF_SPART + wg] = sacc;
}

// ------- final: delta = A*v + bias*sum ; beta += delta ; crit ; flags -------
__global__ void __launch_bounds__(256) k_update(const float* __restrict__ bias,
                                                float* ws, float* out) {
  HubState* st = (HubState*)ws;
  if (st->done) return;
  const int t = threadIdx.x;
  __shared__ float vred[4][64];
  __shared__ float vfin[64];
  __shared__ float sred[256];

  float scale = st->scale;
  float Schi = 0.f;
  for (int i = 0; i < GA_RES; ++i) Schi += ws[OFF_CHI + i];   // same order as k_pseu
  float ns = new_scale(scale, Schi);

  // s = sum(r_pseu)
  float sl = 0.f;
  for (int i = t; i < WAVES_PSEU; i += 256) sl += ws[OFF_SPART + i];
  sred[t] = sl;
  __syncthreads();
  for (int s = 128; s > 0; s >>= 1) { if (t < s) sred[t] += sred[t + s]; __syncthreads(); }
  float ssum = sred[0];
  __syncthreads();

  // v[64] = sum of per-wave partials
  int col = t & 63, grp = t >> 6;
  {
    float acc = 0.f;
    const float* vp = ws + OFF_VPART;
    int w0 = grp * (WAVES_PSEU / 4);
    for (int w = w0; w < w0 + WAVES_PSEU / 4; ++w) acc += vp[(size_t)w * 64 + col];
    vred[grp][col] = acc;
  }
  __syncthreads();
  if (t < 64) vfin[t] = vred[0][t] + vred[1][t] + vred[2][t] + vred[3][t];
  __syncthreads();

  float beta_old = 0.f, delta = 0.f;
  if (t < 64) {
    beta_old = ws[OFF_BETA + t];
    const float* A = ws + OFF_AMAT + t * 64;
    #pragma unroll 8
    for (int k = 0; k < 64; ++k) delta += A[k] * vfin[k];
    delta += bias[t] * ssum;
  }
  sred[t] = delta * delta;
  __syncthreads();
  for (int s = 128; s > 0; s >>= 1) { if (t < s) sred[t] += sred[t + s]; __syncthreads(); }
  float nd = sqrtf(sred[0]);
  __syncthreads();
  sred[t] = beta_old * beta_old;
  __syncthreads();
  for (int s = 128; s > 0; s >>= 1) { if (t < s) sred[t] += sred[t + s]; __syncthreads(); }
  float crit = nd / sqrtf(sred[0]);   // beta0=0 -> inf -> continue (matches jnp)

  if (t < 64) {
    float bn = beta_old + delta;
    ws[OFF_BETA + t] = bn;
    out[t] = bn;
  }
  if (t == 0) {
    st->scale = ns;
    st->crit = crit;
    unsigned c = st->cnt + 1u;
    st->cnt = c;
    if (!(crit >= EPS_C) || c >= (unsigned)MAXITER) st->done = 1u;  // NaN also stops
  }
}

extern "C" void kernel_launch(void* const* d_in, const int* in_sizes, int n_in,
                              void* d_out, int out_size, void* d_ws, size_t ws_size,
                              hipStream_t stream) {
  (void)in_sizes; (void)n_in; (void)out_size; (void)ws_size;
  const float* b    = (const float*)d_in[0];
  const float* X    = (const float*)d_in[1];
  const float* W    = (const float*)d_in[2];
  const float* bias = (const float*)d_in[3];
  float* out = (float*)d_out;
  float* ws  = (float*)d_ws;

  k_init<<<1, 256, 0, stream>>>(ws);

  // scale0 = 1.4815 * lower_median(|b - lower_median(b)|)  (exact radix select)
  for (int m = 0; m < 2; ++m)
    for (int pass = 0; pass < 4; ++pass) {
      k_hist<<<256, 256, 0, stream>>>(b, ws, m, pass);
      k_select<<<1, 1, 0, stream>>>(ws, m, pass);
    }
  k_scale0<<<1, 1, 0, stream>>>(ws);

  // pinv pieces: G = X^T X (WMMA), Ginv, A = W*Ginv  (all loop-invariant)
  k_gram<<<GB_GRAM, 512, 0, stream>>>(X, ws);
  k_gram_reduce<<<16, 256, 0, stream>>>(ws);
  k_invert<<<1, 256, 0, stream>>>(ws);
  k_prepA<<<1, 256, 0, stream>>>(W, ws);

  // IRLS: fixed launch sequence, device-side early exit via done flag
  for (int it = 0; it < MAXITER; ++it) {
    k_resid<<<GA_RES, 256, 0, stream>>>(X, b, ws);
    k_pseu<<<GP_BLK, 256, 0, stream>>>(X, ws);
    k_update<<<1, 256, 0, stream>>>(bias, ws, out);
  }
}